// CRF_31911607009721
// MI455X (gfx1250) — compile-verified
//
#include <hip/hip_runtime.h>
#include <math.h>

typedef float v2f __attribute__((ext_vector_type(2)));
typedef float v8f __attribute__((ext_vector_type(8)));

#define Bv 32
#define Tv 512
#define Lv 64

// ------------------------------------------------------------------
// Kernel 1: path_score[b] = h_score[b] + g_score[b]
//   h = sum_{t,l} inputs[b,t,l] * labels[b,t,l]   (labels are one-hot)
//   g = sum_t trans[idx[t], idx[t+1]]
// ------------------------------------------------------------------
__global__ __launch_bounds__(256)
void crf_path_kernel(const float* __restrict__ inputs,
                     const float* __restrict__ labels,
                     const float* __restrict__ trans,
                     float* __restrict__ path /* [B] */) {
    __shared__ int   sidx[Tv];
    __shared__ float sred[256];
    const int b   = blockIdx.x;
    const int tid = threadIdx.x;
    const float* in_b = inputs + (size_t)b * Tv * Lv;
    const float* lb_b = labels + (size_t)b * Tv * Lv;

    float h = 0.f;
    for (int t = tid; t < Tv; t += 256) {
        const float* lrow = lb_b + t * Lv;
        const float* irow = in_b + t * Lv;
        int   idx = 0;
        float hh  = 0.f;
        for (int l = 0; l < Lv; ++l) {
            float lv = lrow[l];
            hh += irow[l] * lv;
            if (lv > 0.5f) idx = l;       // one-hot -> label index
        }
        h += hh;
        sidx[t] = idx;
    }
    __syncthreads();

    float g = 0.f;
    for (int t = tid; t < Tv - 1; t += 256)
        g += trans[sidx[t] * Lv + sidx[t + 1]];

    sred[tid] = h + g;
    __syncthreads();
    for (int s = 128; s > 0; s >>= 1) {
        if (tid < s) sred[tid] += sred[tid + s];
        __syncthreads();
    }
    if (tid == 0) path[b] = sred[0];
}

// ------------------------------------------------------------------
// Kernel 2: forward recursion via WMMA f32 16x16x4.
// Single workgroup (256 thr = 8 waves). Wave w owns output tile
// (mTile = w>>2, nTile = w&3) of the 32x64 state matrix.
// Per step:  m = rowmax(Z); E = exp(Z-m); S = E @ expT (WMMA);
//            Z = log(S) + m + inputs[:,t,:]
// ------------------------------------------------------------------
__global__ __launch_bounds__(256)
void crf_forward_kernel(const float* __restrict__ inputs,
                        const float* __restrict__ trans,
                        const float* __restrict__ path,
                        float* __restrict__ out) {
    __shared__ float sExpT[Lv * Lv];   // 16 KB, exp(trans), row-major [i][j]
    __shared__ float sZ[Bv * Lv];      // 8 KB
    __shared__ float sE[Bv * Lv];      // 8 KB
    __shared__ float sPMax[Bv][8];
    __shared__ float sM[Bv];
    __shared__ float sLogZ[Bv];

    const int tid      = threadIdx.x;
    const int lane     = tid & 31;
    const int wave     = tid >> 5;      // 0..7
    const int mBase    = (wave >> 2) * 16;   // 0 or 16
    const int nBase    = (wave & 3) * 16;    // 0,16,32,48
    const int laneHalf = lane >> 4;     // 0: lanes 0-15, 1: lanes 16-31
    const int lane15   = lane & 15;

    // exp(trans) into LDS (constant across time)
    for (int i = tid; i < Lv * Lv; i += 256)
        sExpT[i] = __expf(trans[i]);

    // Z0 = inputs[:, 0, :]
    for (int i = tid; i < Bv * Lv; i += 256)
        sZ[i] = inputs[(size_t)(i >> 6) * Tv * Lv + (i & 63)];
    __syncthreads();

    // Resident B fragments: B tile kt is rows [4kt..4kt+3] x cols [nBase..nBase+15]
    // layout: VGPR0 = row 4kt + 2*laneHalf, VGPR1 = row 4kt + 2*laneHalf + 1
    v2f bfrag[16];
    for (int kt = 0; kt < 16; ++kt) {
        int k0 = kt * 4 + laneHalf * 2;
        bfrag[kt].x = sExpT[(k0 + 0) * Lv + nBase + lane15];
        bfrag[kt].y = sExpT[(k0 + 1) * Lv + nBase + lane15];
    }

    for (int t = 1; t < Tv; ++t) {
        // --- stage 1: row maxima (two-level: 8 partials per row) ---
        {
            int b = tid >> 3, c = (tid & 7) * 8;
            float m = sZ[b * Lv + c];
            for (int k = 1; k < 8; ++k) m = fmaxf(m, sZ[b * Lv + c + k]);
            sPMax[b][tid & 7] = m;
        }
        __syncthreads();
        if (tid < Bv) {
            float m = sPMax[tid][0];
            for (int k = 1; k < 8; ++k) m = fmaxf(m, sPMax[tid][k]);
            sM[tid] = m;
        }
        __syncthreads();

        // --- stage 2: E = exp(Z - m) ---
        {
            int b = tid >> 3, c = (tid & 7) * 8;
            float m = sM[b];
            for (int k = 0; k < 8; ++k)
                sE[b * Lv + c + k] = __expf(sZ[b * Lv + c + k] - m);
        }
        __syncthreads();

        // --- stage 3: S = E @ expT via 16 chained WMMA f32 16x16x4 ---
        v8f acc = (v8f){0.f, 0.f, 0.f, 0.f, 0.f, 0.f, 0.f, 0.f};
        {
            int row = mBase + lane15;
            for (int kt = 0; kt < 16; ++kt) {
                int k0 = kt * 4 + laneHalf * 2;
                v2f a;
                a.x = sE[row * Lv + k0];
                a.y = sE[row * Lv + k0 + 1];
                acc = __builtin_amdgcn_wmma_f32_16x16x4_f32(
                    /*neg_a=*/false, a, /*neg_b=*/false, bfrag[kt],
                    /*c_mod=*/(short)0, acc, /*reuse_a=*/false, /*reuse_b=*/false);
            }
        }

        // --- stage 4: Z = log(S) + m + inputs[:, t, :] ---
        // C/D layout: VGPR v -> (M = v + 8*laneHalf, N = lane15)
        {
            int col = nBase + lane15;
            for (int v = 0; v < 8; ++v) {
                int r = mBase + v + laneHalf * 8;
                float z = __logf(acc[v]) + sM[r]
                        + inputs[((size_t)r * Tv + t) * Lv + col];
                sZ[r * Lv + col] = z;
            }
        }
        __syncthreads();
    }

    // --- final: logZ[b] = LSE_j(Z[b,j]); out = sum_b(logZ[b] - path[b]) ---
    if (tid < Bv) {
        float m = sZ[tid * Lv];
        for (int j = 1; j < Lv; ++j) m = fmaxf(m, sZ[tid * Lv + j]);
        float s = 0.f;
        for (int j = 0; j < Lv; ++j) s += __expf(sZ[tid * Lv + j] - m);
        sLogZ[tid] = m + __logf(s);
    }
    __syncthreads();
    if (tid == 0) {
        float tot = 0.f;
        for (int b = 0; b < Bv; ++b) tot += sLogZ[b] - path[b];
        out[0] = tot;
    }
}

// ------------------------------------------------------------------
extern "C" void kernel_launch(void* const* d_in, const int* in_sizes, int n_in,
                              void* d_out, int out_size, void* d_ws, size_t ws_size,
                              hipStream_t stream) {
    const float* inputs = (const float*)d_in[0];   // (B,T,L) f32
    const float* labels = (const float*)d_in[1];   // (B,T,L) f32 one-hot
    const float* trans  = (const float*)d_in[2];   // (L,L)  f32
    float* out  = (float*)d_out;                   // scalar
    float* path = (float*)d_ws;                    // B floats of scratch

    crf_path_kernel<<<Bv, 256, 0, stream>>>(inputs, labels, trans, path);
    crf_forward_kernel<<<1, 256, 0, stream>>>(inputs, trans, path, out);
}